// DependencyProximity_23905787970107
// MI455X (gfx1250) — compile-verified
//
#include <hip/hip_runtime.h>
#include <stdint.h>

// Problem dims (fixed by the reference)
#define BB 64
#define SS 2048
#define DD 512          // floats per row  -> 128 float4 per row
#define ROWS_PER_BLOCK 16
#define ROWS_PER_TILE  8   // 8 waves per 256-thread block, one row per wave

typedef float fvec4 __attribute__((ext_vector_type(4)));
typedef int   ivec4 __attribute__((ext_vector_type(4)));   // builtin's expected pointee

#ifndef __has_builtin
#define __has_builtin(x) 0
#endif

#if defined(__AMDGCN__) && __has_builtin(__builtin_amdgcn_global_load_async_to_lds_b128)
#define USE_ASYNC_LDS 1
#else
#define USE_ASYNC_LDS 0
#endif

#if USE_ASYNC_LDS
typedef __attribute__((address_space(3))) ivec4 lvec4;     // LDS-qualified int4
#endif

// Per-row scalar weight:
//   w = 1 - dep[b,j] / (text_len[b]-aspect_len[b]);  zeroed if j>=text_len or j in [start,end]
__device__ __forceinline__ float row_weight(int row,
                                            const long long* __restrict__ adi,
                                            const long long* __restrict__ text_len,
                                            const long long* __restrict__ aspect_len,
                                            const int* __restrict__ dep) {
    const int b = row >> 11;           // row / SS
    const int j = row & (SS - 1);      // row % SS
    const long long tl = text_len[b];
    const long long st = adi[2 * b];
    const long long en = adi[2 * b + 1];
    const float ctx = (float)(tl - aspect_len[b]);
    float w = 1.0f - (float)dep[row] / ctx;
    const bool zero = ((long long)j >= tl) ||
                      (((long long)j >= st) && ((long long)j <= en));
    return zero ? 0.0f : w;
}

__global__ __launch_bounds__(256) void dp_scale_kernel(
    const float* __restrict__ x,
    const long long* __restrict__ adi,
    const long long* __restrict__ text_len,
    const long long* __restrict__ aspect_len,
    const int* __restrict__ dep,
    float* __restrict__ out)
{
    const int t    = threadIdx.x;
    const int lane = t & 31;
    const int wave = t >> 5;                      // 0..7 : row within tile
    const int rowBase = blockIdx.x * ROWS_PER_BLOCK;

    const int row0 = rowBase + wave;                   // tile 0 row for this wave
    const int row1 = rowBase + ROWS_PER_TILE + wave;   // tile 1 row for this wave

    const fvec4* __restrict__ src0 = (const fvec4*)(x + (size_t)row0 * DD);
    const fvec4* __restrict__ src1 = (const fvec4*)(x + (size_t)row1 * DD);
    fvec4* __restrict__ dst0 = (fvec4*)(out + (size_t)row0 * DD);
    fvec4* __restrict__ dst1 = (fvec4*)(out + (size_t)row1 * DD);

#if USE_ASYNC_LDS
    // 2 tiles x 8 rows x 128 float4 = 32 KB of LDS, double-buffered async pipeline
    __shared__ fvec4 smem[2][ROWS_PER_TILE * 128];

    // Issue tile 0 async copies (4 x b128 per lane -> 4 ASYNC ops per wave)
    #pragma unroll
    for (int k = 0; k < 4; ++k) {
        const int idx = lane + k * 32;
        __builtin_amdgcn_global_load_async_to_lds_b128(
            (ivec4*)(src0 + idx),
            (lvec4*)(&smem[0][wave * 128 + idx]), 0, 0);
    }
    // Issue tile 1 async copies (4 more ASYNC ops per wave)
    #pragma unroll
    for (int k = 0; k < 4; ++k) {
        const int idx = lane + k * 32;
        __builtin_amdgcn_global_load_async_to_lds_b128(
            (ivec4*)(src1 + idx),
            (lvec4*)(&smem[1][wave * 128 + idx]), 0, 0);
    }

    // Overlap: compute tile-0 weight while both tiles are in flight
    const float w0 = row_weight(row0, adi, text_len, aspect_len, dep);

    // Wait until the first 4 async ops (tile 0) have landed; tile 1 may still fly
    asm volatile("s_wait_asynccnt 0x4" ::: "memory");

    #pragma unroll
    for (int k = 0; k < 4; ++k) {
        const int idx = lane + k * 32;
        fvec4 v = smem[0][wave * 128 + idx];
        v *= w0;
        __builtin_nontemporal_store(v, dst0 + idx);   // NT: no reuse, keep L2 clean
    }

    const float w1 = row_weight(row1, adi, text_len, aspect_len, dep);

    asm volatile("s_wait_asynccnt 0x0" ::: "memory");

    #pragma unroll
    for (int k = 0; k < 4; ++k) {
        const int idx = lane + k * 32;
        fvec4 v = smem[1][wave * 128 + idx];
        v *= w1;
        __builtin_nontemporal_store(v, dst1 + idx);
    }
#else
    // Fallback: direct wide NT loads, still fully coalesced b128 traffic
    const float w0 = row_weight(row0, adi, text_len, aspect_len, dep);
    #pragma unroll
    for (int k = 0; k < 4; ++k) {
        const int idx = lane + k * 32;
        fvec4 v = __builtin_nontemporal_load(src0 + idx);
        v *= w0;
        __builtin_nontemporal_store(v, dst0 + idx);
    }
    const float w1 = row_weight(row1, adi, text_len, aspect_len, dep);
    #pragma unroll
    for (int k = 0; k < 4; ++k) {
        const int idx = lane + k * 32;
        fvec4 v = __builtin_nontemporal_load(src1 + idx);
        v *= w1;
        __builtin_nontemporal_store(v, dst1 + idx);
    }
#endif
}

extern "C" void kernel_launch(void* const* d_in, const int* in_sizes, int n_in,
                              void* d_out, int out_size, void* d_ws, size_t ws_size,
                              hipStream_t stream) {
    (void)in_sizes; (void)n_in; (void)out_size; (void)d_ws; (void)ws_size;
    const float*     x   = (const float*)d_in[0];       // (B,S,D) f32
    const long long* adi = (const long long*)d_in[1];   // (B,2)  i64
    const long long* tl  = (const long long*)d_in[2];   // (B,)   i64
    const long long* al  = (const long long*)d_in[3];   // (B,)   i64
    const int*       dep = (const int*)d_in[4];         // (B,S)  i32
    float* out = (float*)d_out;                         // (B,S,D) f32

    const int totalRows = BB * SS;                      // 131072
    dim3 grid(totalRows / ROWS_PER_BLOCK);              // 8192 blocks
    dim3 block(256);                                    // 8 waves (wave32)
    dp_scale_kernel<<<grid, block, 0, stream>>>(x, adi, tl, al, dep, out);
}